// TradingOptimizationLoss_22866405884191
// MI455X (gfx1250) — compile-verified
//
#include <hip/hip_runtime.h>
#include <stdint.h>

#define TPB 256
#define EPT 16
#define CHUNK (TPB * EPT)        // 4096 elements per block
#define NWAVES (TPB / 32)        // 8 waves (wave32)
#define WTILE (CHUNK / NWAVES)   // 512 elements per wave tile

typedef unsigned int u32x4 __attribute__((ext_vector_type(4)));
typedef int          i32x4 __attribute__((ext_vector_type(4)));
typedef int          i32x8 __attribute__((ext_vector_type(8)));

#if __has_builtin(__builtin_amdgcn_tensor_load_to_lds) && __has_builtin(__builtin_amdgcn_s_wait_tensorcnt)
#define USE_TDM 1
#else
#define USE_TDM 0
#endif

#if USE_TDM
// 1D TDM load: nvalid fp32 elements (OOB reads beyond tensor_dim0 return 0),
// tile width = `tile` elements, into LDS at byte address ldsb.
__device__ __forceinline__ void tdm_load_f32(const float* gsrc, uint32_t ldsb,
                                             uint32_t nvalid, uint32_t tile) {
  uint64_t ga = (uint64_t)(uintptr_t)gsrc;
  u32x4 g0;
  g0[0] = 1u;                                              // count=1 valid descriptor
  g0[1] = ldsb;                                            // lds_addr [63:32]
  g0[2] = (uint32_t)ga;                                    // global_addr[31:0]
  g0[3] = ((uint32_t)(ga >> 32) & 0x01FFFFFFu) | (2u << 30); // addr[56:32] | type=2
  i32x8 g1;
  g1[0] = (int)(2u << 16);                                 // data_size = 4 bytes
  g1[1] = (int)((nvalid & 0xFFFFu) << 16);                 // tensor_dim0[15:0]
  g1[2] = (int)(((nvalid >> 16) & 0xFFFFu) | (1u << 16));  // dim0[31:16] | tensor_dim1=1
  g1[3] = (int)((tile & 0xFFFFu) << 16);                   // tile_dim0
  g1[4] = 0;                                               // tile_dim1/2 unused
  g1[5] = (int)tile;                                       // tensor_dim0_stride (unused for 1D)
  g1[6] = 0;
  g1[7] = 0;
  i32x4 z4 = {0, 0, 0, 0};
  i32x8 z8 = {0, 0, 0, 0, 0, 0, 0, 0};
  __builtin_amdgcn_tensor_load_to_lds(g0, g1, z4, z4, z8, 0);
}
#endif

// Load a CHUNK of positions+returns into LDS (zero-filled past T).
__device__ __forceinline__ void load_tile(const float* __restrict__ gp,
                                          const float* __restrict__ gr,
                                          float* s_pos, float* s_ret,
                                          long base, int T) {
#if USE_TDM
  int wave = __builtin_amdgcn_readfirstlane((int)(threadIdx.x >> 5));
  int sub = wave * WTILE;
  long g = base + (long)sub;
  long rem = (long)T - g;
  if (rem > 0) {
    uint32_t nv = (uint32_t)(rem < (long)WTILE ? rem : (long)WTILE);
    tdm_load_f32(gp + g, (uint32_t)(uintptr_t)(s_pos + sub), nv, WTILE);
    tdm_load_f32(gr + g, (uint32_t)(uintptr_t)(s_ret + sub), nv, WTILE);
  }
  __builtin_amdgcn_s_wait_tensorcnt(0);
  __syncthreads();
#else
  for (int i = threadIdx.x; i < CHUNK; i += TPB) {
    long g = base + i;
    bool v = g < (long)T;
    s_pos[i] = v ? gp[g] : 0.0f;
    s_ret[i] = v ? gr[g] : 0.0f;
  }
  __syncthreads();
#endif
}

// ---- block-wide LDS scan/reduce primitives (256 threads = 8 wave32) ----
__device__ __forceinline__ float blk_scan_incl_sum(float v, float* s, int t) {
  __syncthreads(); s[t] = v; __syncthreads();
  #pragma unroll
  for (int off = 1; off < TPB; off <<= 1) {
    float o = (t >= off) ? s[t - off] : 0.0f;
    __syncthreads();
    s[t] += o;
    __syncthreads();
  }
  return s[t];
}
__device__ __forceinline__ float blk_scan_incl_max(float v, float* s, int t) {
  __syncthreads(); s[t] = v; __syncthreads();
  #pragma unroll
  for (int off = 1; off < TPB; off <<= 1) {
    float o = (t >= off) ? s[t - off] : -INFINITY;
    __syncthreads();
    s[t] = fmaxf(s[t], o);
    __syncthreads();
  }
  return s[t];
}
__device__ __forceinline__ float blk_reduce_sum(float v, float* s, int t) {
  __syncthreads(); s[t] = v; __syncthreads();
  #pragma unroll
  for (int w = TPB >> 1; w > 0; w >>= 1) { if (t < w) s[t] += s[t + w]; __syncthreads(); }
  return s[0];
}
__device__ __forceinline__ float blk_reduce_max(float v, float* s, int t) {
  __syncthreads(); s[t] = v; __syncthreads();
  #pragma unroll
  for (int w = TPB >> 1; w > 0; w >>= 1) { if (t < w) s[t] = fmaxf(s[t], s[t + w]); __syncthreads(); }
  return s[0];
}

__device__ __forceinline__ float clip2(float x) {
  return fminf(fmaxf(x, -2.0f), 2.0f);
}

// ---------- Pass 1: per-block partials ----------
__global__ void __launch_bounds__(TPB)
k_pass1(const float* __restrict__ pos, const float* __restrict__ ret,
        float* w_total, float* w_lmax, float* w_s1, float* w_s2,
        float* w_sd2, float* w_ad, int T) {
  __shared__ float s_pos[CHUNK];
  __shared__ float s_ret[CHUNK];
  __shared__ float s_red[TPB];

  int b = blockIdx.x;
  int t = threadIdx.x;
  long base = (long)b * CHUNK;
  load_tile(pos, ret, s_pos, s_ret, base, T);

  float pr[EPT], pcl[EPT];
  #pragma unroll
  for (int k = 0; k < EPT; ++k) {
    float pc = clip2(s_pos[t * EPT + k]);
    pcl[k] = pc;
    pr[k] = pc * s_ret[t * EPT + k];
  }
  // element following this thread's run (for diff terms)
  float pr_next = 0.0f, pcl_next = 0.0f;
  {
    long gnext = base + (long)(t + 1) * EPT;
    if (t < TPB - 1) {
      float pc = clip2(s_pos[(t + 1) * EPT]);
      pcl_next = pc; pr_next = pc * s_ret[(t + 1) * EPT];
    } else if (gnext < (long)T) {
      float pc = clip2(pos[gnext]);
      pcl_next = pc; pr_next = pc * ret[gnext];
    }
  }

  float s1 = 0.f, s2 = 0.f, sd2 = 0.f, ad = 0.f, c = 0.f, lmax = -INFINITY;
  #pragma unroll
  for (int k = 0; k < EPT; ++k) {
    long gi = base + (long)t * EPT + k;
    bool v = gi < (long)T;
    float x = v ? pr[k] : 0.0f;
    s1 += x; s2 += x * x; c += x;
    if (v) lmax = fmaxf(lmax, c);
    if (gi < (long)T - 1) {
      float nx = (k < EPT - 1) ? pr[k + 1] : pr_next;
      float nc = (k < EPT - 1) ? pcl[k + 1] : pcl_next;
      float d = nx - x;
      sd2 += d * d;
      ad += fabsf(nc - pcl[k]);
    }
  }

  float incl = blk_scan_incl_sum(c, s_red, t);
  float excl = incl - c;
  float btotal = s_red[TPB - 1];
  float bmax = blk_reduce_max(excl + lmax, s_red, t);
  float S1 = blk_reduce_sum(s1, s_red, t);
  float S2 = blk_reduce_sum(s2, s_red, t);
  float SD2 = blk_reduce_sum(sd2, s_red, t);
  float AD = blk_reduce_sum(ad, s_red, t);
  if (t == 0) {
    w_total[b] = btotal; w_lmax[b] = bmax;
    w_s1[b] = S1; w_s2[b] = S2; w_sd2[b] = SD2; w_ad[b] = AD;
  }
}

// ---------- Pass 2: scan block records (single block) ----------
__global__ void __launch_bounds__(TPB)
k_pass2(const float* w_total, const float* w_lmax,
        const float* w_s1, const float* w_s2, const float* w_sd2, const float* w_ad,
        float* w_prefix, float* w_prefmax, double* w_scal, int NB) {
  __shared__ float s_red[TPB];
  __shared__ double s_redd[TPB];
  int t = threadIdx.x;
  int per = (NB + TPB - 1) / TPB;
  int b0 = t * per;

  float tsum = 0.0f;
  for (int i = 0; i < per; ++i) { int b = b0 + i; if (b < NB) tsum += w_total[b]; }
  float incl = blk_scan_incl_sum(tsum, s_red, t);
  float texcl = incl - tsum;

  // write per-block exclusive prefixes; track this thread's max cumsum candidate
  float run = texcl, tcmax = -INFINITY;
  for (int i = 0; i < per; ++i) {
    int b = b0 + i;
    if (b < NB) {
      w_prefix[b] = run;
      tcmax = fmaxf(tcmax, run + w_lmax[b]);
      run += w_total[b];
    }
  }
  blk_scan_incl_max(tcmax, s_red, t);
  float rbase = (t > 0) ? fmaxf(0.0f, s_red[t - 1]) : 0.0f;  // includes max(.,0) clip
  float runm = rbase;
  for (int i = 0; i < per; ++i) {
    int b = b0 + i;
    if (b < NB) {
      w_prefmax[b] = runm;
      runm = fmaxf(runm, w_prefix[b] + w_lmax[b]);
    }
  }

  double a1 = 0, a2 = 0, a3 = 0, a4 = 0;
  for (int i = 0; i < per; ++i) {
    int b = b0 + i;
    if (b < NB) { a1 += w_s1[b]; a2 += w_s2[b]; a3 += w_sd2[b]; a4 += w_ad[b]; }
  }
  __syncthreads(); s_redd[t] = a1; __syncthreads();
  for (int w = TPB >> 1; w > 0; w >>= 1) { if (t < w) s_redd[t] += s_redd[t + w]; __syncthreads(); }
  double S1 = s_redd[0];
  __syncthreads(); s_redd[t] = a2; __syncthreads();
  for (int w = TPB >> 1; w > 0; w >>= 1) { if (t < w) s_redd[t] += s_redd[t + w]; __syncthreads(); }
  double S2 = s_redd[0];
  __syncthreads(); s_redd[t] = a3; __syncthreads();
  for (int w = TPB >> 1; w > 0; w >>= 1) { if (t < w) s_redd[t] += s_redd[t + w]; __syncthreads(); }
  double S3 = s_redd[0];
  __syncthreads(); s_redd[t] = a4; __syncthreads();
  for (int w = TPB >> 1; w > 0; w >>= 1) { if (t < w) s_redd[t] += s_redd[t + w]; __syncthreads(); }
  double S4 = s_redd[0];
  if (t == 0) { w_scal[0] = S1; w_scal[1] = S2; w_scal[2] = S3; w_scal[3] = S4; }
}

// ---------- Pass 3: drawdowns (re-streams from L2-resident data) ----------
__global__ void __launch_bounds__(TPB)
k_pass3(const float* __restrict__ pos, const float* __restrict__ ret,
        const float* w_prefix, const float* w_prefmax,
        float* w_ddsum, float* w_ddmax, int T) {
  __shared__ float s_pos[CHUNK];
  __shared__ float s_ret[CHUNK];
  __shared__ float s_red[TPB];

  int b = blockIdx.x;
  int t = threadIdx.x;
  long base = (long)b * CHUNK;
  load_tile(pos, ret, s_pos, s_ret, base, T);

  float ck[EPT];
  float c = 0.f, lmax = -INFINITY;
  #pragma unroll
  for (int k = 0; k < EPT; ++k) {
    long gi = base + (long)t * EPT + k;
    bool v = gi < (long)T;
    float x = v ? clip2(s_pos[t * EPT + k]) * s_ret[t * EPT + k] : 0.0f;
    c += x;
    ck[k] = c;
    if (v) lmax = fmaxf(lmax, c);
  }

  float incl = blk_scan_incl_sum(c, s_red, t);
  float excl = incl - c;
  float prefix = w_prefix[b];
  float prefmax = w_prefmax[b];

  blk_scan_incl_max(prefix + excl + lmax, s_red, t);
  float rbase = (t > 0) ? fmaxf(prefmax, s_red[t - 1]) : prefmax;

  float ddsum = 0.0f, ddmax = 0.0f, runm = rbase;
  #pragma unroll
  for (int k = 0; k < EPT; ++k) {
    long gi = base + (long)t * EPT + k;
    if (gi < (long)T) {
      float g = prefix + excl + ck[k];
      runm = fmaxf(runm, g);           // running_max = max(cummax, 0) via seed
      float dd = runm - g;
      ddsum += dd;
      ddmax = fmaxf(ddmax, dd);
    }
  }
  float DS = blk_reduce_sum(ddsum, s_red, t);
  float DM = blk_reduce_max(ddmax, s_red, t);
  if (t == 0) { w_ddsum[b] = DS; w_ddmax[b] = DM; }
}

// ---------- Pass 4: final composition ----------
__global__ void __launch_bounds__(TPB)
k_pass4(const float* __restrict__ pos, const float* __restrict__ ret,
        const float* w_ddsum, const float* w_ddmax, const double* w_scal,
        float* out, int T, int NB) {
  __shared__ double sd[TPB];
  __shared__ float sf[TPB];
  int t = threadIdx.x;
  double a = 0.0; float m = 0.0f;
  for (int b = t; b < NB; b += TPB) { a += (double)w_ddsum[b]; m = fmaxf(m, w_ddmax[b]); }
  sd[t] = a; sf[t] = m; __syncthreads();
  for (int w = TPB >> 1; w > 0; w >>= 1) {
    if (t < w) { sd[t] += sd[t + w]; sf[t] = fmaxf(sf[t], sf[t + w]); }
    __syncthreads();
  }
  if (t == 0) {
    double DDSUM = sd[0], DDMAX = (double)sf[0];
    double S1 = w_scal[0], S2 = w_scal[1], SD2 = w_scal[2], AD = w_scal[3];
    double n = (double)T;
    double mean = S1 / n;
    double var = (S2 - S1 * S1 / n) / (n - 1.0); if (var < 0.0) var = 0.0;
    double return_loss = -(mean / (sqrt(var) + 1e-6));
    double dd_loss = DDMAX + 0.5 * (DDSUM / n);
    double pr0 = (double)(clip2(pos[0]) * ret[0]);
    double prL = (double)(clip2(pos[T - 1]) * ret[T - 1]);
    double nd = n - 1.0;
    double md = (prL - pr0) / nd;                      // telescoped mean of diffs
    double vard = (SD2 - nd * md * md) / (nd - 1.0); if (vard < 0.0) vard = 0.0;
    double smooth = sqrt(vard);
    double pvol = AD / nd;
    out[0] = (float)(return_loss + dd_loss + 0.5 * smooth + 0.2 * pvol);
  }
}

extern "C" void kernel_launch(void* const* d_in, const int* in_sizes, int n_in,
                              void* d_out, int out_size, void* d_ws, size_t ws_size,
                              hipStream_t stream) {
  (void)n_in; (void)out_size; (void)ws_size;
  const float* pos = (const float*)d_in[0];
  const float* ret = (const float*)d_in[1];
  int T = in_sizes[0];
  int NB = (T + CHUNK - 1) / CHUNK;

  float* ws = (float*)d_ws;
  float* w_total   = ws + 0L * NB;
  float* w_lmax    = ws + 1L * NB;
  float* w_s1      = ws + 2L * NB;
  float* w_s2      = ws + 3L * NB;
  float* w_sd2     = ws + 4L * NB;
  float* w_ad      = ws + 5L * NB;
  float* w_prefix  = ws + 6L * NB;
  float* w_prefmax = ws + 7L * NB;
  float* w_ddsum   = ws + 8L * NB;
  float* w_ddmax   = ws + 9L * NB;
  double* w_scal = (double*)(((uintptr_t)(ws + 10L * NB) + 7) & ~(uintptr_t)7);

  k_pass1<<<NB, TPB, 0, stream>>>(pos, ret, w_total, w_lmax, w_s1, w_s2, w_sd2, w_ad, T);
  k_pass2<<<1, TPB, 0, stream>>>(w_total, w_lmax, w_s1, w_s2, w_sd2, w_ad,
                                 w_prefix, w_prefmax, w_scal, NB);
  k_pass3<<<NB, TPB, 0, stream>>>(pos, ret, w_prefix, w_prefmax, w_ddsum, w_ddmax, T);
  k_pass4<<<1, TPB, 0, stream>>>(pos, ret, w_ddsum, w_ddmax, w_scal, (float*)d_out, T, NB);
}